// DGCNN_16423954940541
// MI455X (gfx1250) — compile-verified
//
#include <hip/hip_runtime.h>
#include <math.h>

typedef float v2f __attribute__((ext_vector_type(2)));
typedef float v8f __attribute__((ext_vector_type(8)));

#define B_    2
#define N_    512
#define KNN_  25
#define M_    8
#define PAIRS 300      /* 25*24/2 */
#define PPAD  304      /* padded to multiple of 16 */
#define NT_   (PPAD/16)

// ---------------------------------------------------------------------------
// CDNA5 f32 WMMA 16x16x4:  D = A(16x4) * B(4x16) + C(16x16)
// A frag (2 VGPR): lane l: row M = l&15 ; K pair = (l>=16)?{2,3}:{0,1}
// B frag (2 VGPR): lane l: col N = l&15 ; K pair = (l>=16)?{2,3}:{0,1}
// D frag (8 VGPR): lane l, vgpr v: M = v + 8*(l>=16), N = l&15
// ---------------------------------------------------------------------------
__device__ __forceinline__ v8f wmma4(v2f a, v2f b, v8f c) {
  return __builtin_amdgcn_wmma_f32_16x16x4_f32(false, a, false, b, (short)0, c,
                                               false, false);
}

__device__ __forceinline__ float gelu_f(float v) {
  return 0.5f * v * (1.0f + erff(v * 0.70710678118654752440f));
}

// ---------------- xx[i] = sum_c x[c][i]^2 -----------------------------------
__global__ void k_xx(const float* __restrict__ x, float* __restrict__ xx, int C) {
  int i = blockIdx.x * blockDim.x + threadIdx.x;
  if (i >= B_ * N_) return;
  int b = i / N_, n = i % N_;
  const float* xb = x + (size_t)b * C * N_;
  float s = 0.f;
  for (int c = 0; c < C; ++c) { float v = xb[(size_t)c * N_ + n]; s += v * v; }
  xx[i] = s;
}

// --------- neg_d2 = 2*(x^T x) - xx_i - xx_j via f32 WMMA Gram ---------------
// Input has exactly CPAD channels (pad channels zeroed). One wave computes a
// 16x64 strip: A fragment reused across 4 column tiles.
template <int CPAD>
__global__ void k_gram_negd2(const float* __restrict__ x,
                             const float* __restrict__ xx,
                             float* __restrict__ nd2) {
  const int TI = N_ / 16;       // 32 row tiles
  const int JG = N_ / 64;      // 8 groups of 4 col tiles
  int t = blockIdx.x;           // b*TI*JG + ti*JG + jg
  int b = t / (TI * JG);
  int r = t % (TI * JG);
  int ti = r / JG, jg = r % JG;
  int lane = threadIdx.x & 31;
  int half = lane >> 4, lidx = lane & 15;
  int i0 = ti * 16, j0 = jg * 64;
  const float* xb = x + (size_t)b * CPAD * N_;
  v8f acc0 = {}, acc1 = {}, acc2 = {}, acc3 = {};
#pragma unroll 4
  for (int c0 = 0; c0 < CPAD; c0 += 4) {
    int k = c0 + half * 2;
    const float* r0 = xb + (size_t)k * N_;
    const float* r1 = r0 + N_;
    v2f a;  a.x  = r0[i0 + lidx];      a.y  = r1[i0 + lidx];
    v2f b0; b0.x = r0[j0 + lidx];      b0.y = r1[j0 + lidx];
    v2f b1; b1.x = r0[j0 + 16 + lidx]; b1.y = r1[j0 + 16 + lidx];
    v2f b2; b2.x = r0[j0 + 32 + lidx]; b2.y = r1[j0 + 32 + lidx];
    v2f b3; b3.x = r0[j0 + 48 + lidx]; b3.y = r1[j0 + 48 + lidx];
    acc0 = wmma4(a, b0, acc0);
    acc1 = wmma4(a, b1, acc1);
    acc2 = wmma4(a, b2, acc2);
    acc3 = wmma4(a, b3, acc3);
  }
  const float* xxb = xx + b * N_;
  float* outb = nd2 + (size_t)b * N_ * N_;
  v8f accs[4] = {acc0, acc1, acc2, acc3};
#pragma unroll
  for (int jt = 0; jt < 4; ++jt) {
    float xj = xxb[j0 + jt * 16 + lidx];
#pragma unroll
    for (int v = 0; v < 8; ++v) {
      int m = v + 8 * half;
      outb[(size_t)(i0 + m) * N_ + j0 + jt * 16 + lidx] =
          2.f * accs[jt][v] - xxb[i0 + m] - xj;
    }
  }
}

// --------- top-25 neighbor indices per row (largest neg_d2) -----------------
__global__ void k_topk(const float* __restrict__ nd2, int* __restrict__ idx) {
  int row = blockIdx.x * blockDim.x + threadIdx.x;   // over B*N
  if (row >= B_ * N_) return;
  const float* d = nd2 + (size_t)row * N_;
  float val[KNN_]; int vid[KNN_];
  for (int t = 0; t < KNN_; ++t) { val[t] = -INFINITY; vid[t] = 0; }
  for (int j = 0; j < N_; ++j) {
    float v = d[j];
    if (v > val[KNN_ - 1]) {
      int p = KNN_ - 1;
      while (p > 0 && v > val[p - 1]) { val[p] = val[p - 1]; vid[p] = vid[p - 1]; --p; }
      val[p] = v; vid[p] = j;
    }
  }
  int* o = idx + (size_t)row * KNN_;
  for (int t = 0; t < KNN_; ++t) o[t] = vid[t];
}

// --------- per-point normal voting: cross products + WMMA sim Gram ----------
__global__ void k_vecavg(const float* __restrict__ x, int Cstride,
                         const int* __restrict__ idx, float* __restrict__ nout) {
  __shared__ float vec[KNN_][3];
  __shared__ float unitv[PPAD][4];
  __shared__ float simv[PPAD];
  __shared__ unsigned char validv[PPAD];
  int pt = blockIdx.x;
  int b = pt / N_, n = pt % N_;
  const float* xb = x + (size_t)b * Cstride * N_;
  int tid = threadIdx.x;

  if (tid < KNN_) {
    int j = idx[(size_t)pt * KNN_ + tid];
#pragma unroll
    for (int c = 0; c < 3; ++c)
      vec[tid][c] = xb[(size_t)c * N_ + j] - xb[(size_t)c * N_ + n];
  }
  __syncthreads();

  for (int p = tid; p < PPAD; p += blockDim.x) {
    if (p < PAIRS) {
      int i = 0, rem = p;
      while (rem >= (KNN_ - 1 - i)) { rem -= (KNN_ - 1 - i); ++i; }
      int j = i + 1 + rem;
      float ax = vec[i][0], ay = vec[i][1], az = vec[i][2];
      float bx = vec[j][0], by = vec[j][1], bz = vec[j][2];
      float cx = ay * bz - az * by;
      float cy = az * bx - ax * bz;
      float cz = ax * by - ay * bx;
      float sq = cx * cx + cy * cy + cz * cz;
      float nr = sqrtf(fmaxf(sq, 1e-24f));
      bool vld = nr > 1e-6f;
      float inv = vld ? 1.f / nr : 0.f;
      unitv[p][0] = cx * inv; unitv[p][1] = cy * inv;
      unitv[p][2] = cz * inv; unitv[p][3] = 0.f;
      validv[p] = vld ? 1 : 0;
    } else {
      unitv[p][0] = unitv[p][1] = unitv[p][2] = unitv[p][3] = 0.f;
      validv[p] = 0;
    }
  }
  __syncthreads();

  // sim_p = sum_q |u_p . u_q| ; 16x16 dot-tiles via one 16x16x4 f32 WMMA each.
  // Keep register pressure low: exactly one A-frag, one B-frag, one D-tile and
  // the 8 row accumulators live at any time (no unrolling -> no spills).
  int wave = tid >> 5, lane = tid & 31;
  int half = lane >> 4, lidx = lane & 15;
#pragma unroll 1
  for (int rt = wave; rt < NT_; rt += 8) {
    int k0 = half * 2;
    int arow = rt * 16 + lidx;
    v2f a; a.x = unitv[arow][k0]; a.y = unitv[arow][k0 + 1];
    float racc[8];
#pragma unroll
    for (int v = 0; v < 8; ++v) racc[v] = 0.f;
#pragma unroll 1
    for (int ct = 0; ct < NT_; ++ct) {
      int bcol = ct * 16 + lidx;
      v2f bb; bb.x = unitv[bcol][k0]; bb.y = unitv[bcol][k0 + 1];
      v8f cz = {};
      v8f d = wmma4(a, bb, cz);
#pragma unroll
      for (int v = 0; v < 8; ++v) racc[v] += fabsf(d[v]);
    }
#pragma unroll
    for (int v = 0; v < 8; ++v) {
      float s = racc[v];
      for (int off = 1; off < 16; off <<= 1) s += __shfl_xor(s, off, 16);
      if (lidx == 0) simv[rt * 16 + v + 8 * half] = s;
    }
  }
  __syncthreads();
  if (tid < PPAD && (tid >= PAIRS || !validv[tid])) simv[tid] = -INFINITY;
  if (tid + 256 < PPAD && ((tid + 256) >= PAIRS || !validv[tid + 256]))
    simv[tid + 256] = -INFINITY;
  __syncthreads();

  if (tid == 0) {
    float bv[M_]; int bi[M_];
    for (int t = 0; t < M_; ++t) { bv[t] = -INFINITY; bi[t] = 0; }
    for (int p = 0; p < PAIRS; ++p) {
      float v = simv[p];
      if (v > bv[M_ - 1]) {
        int q = M_ - 1;
        while (q > 0 && v > bv[q - 1]) { bv[q] = bv[q - 1]; bi[q] = bi[q - 1]; --q; }
        bv[q] = v; bi[q] = p;
      }
    }
    float sx = 0.f, sy = 0.f, sz = 0.f;
    for (int t = 0; t < M_; ++t) {
      sx += unitv[bi[t]][0]; sy += unitv[bi[t]][1]; sz += unitv[bi[t]][2];
    }
    sx *= (1.f / M_); sy *= (1.f / M_); sz *= (1.f / M_);
    float inv = 1.f / (sqrtf(sx * sx + sy * sy + sz * sz) + 1e-6f);
    float* o = nout + (size_t)b * 3 * N_;
    o[0 * N_ + n] = sx * inv; o[1 * N_ + n] = sy * inv; o[2 * N_ + n] = sz * inv;
  }
}

// ---------------- buffer utilities ------------------------------------------
__global__ void k_zero(float* p, int count) {
  int i = blockIdx.x * blockDim.x + threadIdx.x;
  if (i < count) p[i] = 0.f;
}

__global__ void k_pack(float* __restrict__ dst, int dstC, int off,
                       const float* __restrict__ src, int srcC, int nch) {
  int i = blockIdx.x * blockDim.x + threadIdx.x;
  int tot = B_ * nch * N_;
  if (i >= tot) return;
  int n = i % N_;
  int c = (i / N_) % nch;
  int b = i / (N_ * nch);
  dst[((size_t)b * dstC + off + c) * N_ + n] =
      src[((size_t)b * srcC + c) * N_ + n];
}

// zero-pad weights: dst[o][c] = (c < Cin) ? W[o][c] : 0, dst row stride CPad
__global__ void k_padw(float* __restrict__ dst, const float* __restrict__ src,
                       int O, int Cin, int CPad) {
  int i = blockIdx.x * blockDim.x + threadIdx.x;
  if (i >= O * CPad) return;
  int o = i / CPad, c = i % CPad;
  dst[i] = (c < Cin) ? src[(size_t)o * Cin + c] : 0.f;
}

// ---------------- conv GEMM: y[b][o][n] = sum_c Wp[o][c]*in[b][c][n] --------
// Wp is O x CPAD (zero padded), in is B x CPAD x N (pad channels zeroed).
// One wave computes a 16x64 strip (A reused across 4 N-tiles).
template <int CPAD>
__global__ void k_gemm_wmma(const float* __restrict__ Wp,
                            const float* __restrict__ in,
                            float* __restrict__ out, int O) {
  const int NG = N_ / 64;      // 8
  int TO = O / 16;
  int t = blockIdx.x;
  int b = t / (TO * NG);
  int r = t % (TO * NG);
  int to = r / NG, ng = r % NG;
  int lane = threadIdx.x & 31;
  int half = lane >> 4, lidx = lane & 15;
  int o0 = to * 16, n0 = ng * 64;
  const float* inb = in + (size_t)b * CPAD * N_;
  v8f acc0 = {}, acc1 = {}, acc2 = {}, acc3 = {};
#pragma unroll 4
  for (int c0 = 0; c0 < CPAD; c0 += 4) {
    int k = c0 + half * 2;
    v2f a = *(const v2f*)(Wp + (size_t)(o0 + lidx) * CPAD + k);   // b64 load
    const float* r0 = inb + (size_t)k * N_ + n0;
    const float* r1 = r0 + N_;
    v2f b0; b0.x = r0[lidx];      b0.y = r1[lidx];
    v2f b1; b1.x = r0[16 + lidx]; b1.y = r1[16 + lidx];
    v2f b2; b2.x = r0[32 + lidx]; b2.y = r1[32 + lidx];
    v2f b3; b3.x = r0[48 + lidx]; b3.y = r1[48 + lidx];
    if (c0 + 4 < CPAD)             // wave-uniform scalar branch
      __builtin_prefetch(inb + (size_t)(c0 + 4) * N_ + n0, 0, 1);
    acc0 = wmma4(a, b0, acc0);
    acc1 = wmma4(a, b1, acc1);
    acc2 = wmma4(a, b2, acc2);
    acc3 = wmma4(a, b3, acc3);
  }
  float* ob = out + (size_t)b * O * N_;
  v8f accs[4] = {acc0, acc1, acc2, acc3};
#pragma unroll
  for (int jt = 0; jt < 4; ++jt)
#pragma unroll
    for (int v = 0; v < 8; ++v) {
      int m = v + 8 * half;
      ob[(size_t)(o0 + m) * N_ + n0 + jt * 16 + lidx] = accs[jt][v];
    }
}

// ---------------- BN (batch stats over B*N) + GELU --------------------------
__global__ void k_bnstats(const float* __restrict__ y, float* __restrict__ mu,
                          float* __restrict__ rsig, int C) {
  int c = blockIdx.x;
  __shared__ float ssum[256], ssq[256];
  float s = 0.f, q = 0.f;
  for (int i = threadIdx.x; i < B_ * N_; i += blockDim.x) {
    int b = i / N_, n = i % N_;
    float v = y[((size_t)b * C + c) * N_ + n];
    s += v; q += v * v;
  }
  ssum[threadIdx.x] = s; ssq[threadIdx.x] = q;
  __syncthreads();
  for (int st = 128; st > 0; st >>= 1) {
    if (threadIdx.x < st) {
      ssum[threadIdx.x] += ssum[threadIdx.x + st];
      ssq[threadIdx.x]  += ssq[threadIdx.x + st];
    }
    __syncthreads();
  }
  if (threadIdx.x == 0) {
    float m = ssum[0] / (float)(B_ * N_);
    float v = ssq[0] / (float)(B_ * N_) - m * m;
    mu[c] = m;
    rsig[c] = rsqrtf(fmaxf(v, 0.f) + 1e-5f);
  }
}

__global__ void k_bngelu(const float* __restrict__ y, const float* __restrict__ mu,
                         const float* __restrict__ rsig, const float* __restrict__ g,
                         const float* __restrict__ bb, float* __restrict__ dst, int C) {
  int i = blockIdx.x * blockDim.x + threadIdx.x;
  int tot = B_ * C * N_;
  if (i >= tot) return;
  int n = i % N_;
  int c = (i / N_) % C;
  int b = i / (N_ * C);
  size_t o = ((size_t)b * C + c) * N_ + n;
  float v = g[c] * (y[o] - mu[c]) * rsig[c] + bb[c];
  dst[o] = gelu_f(v);
}

// ---------------- max pool over N -------------------------------------------
__global__ void k_maxpool(const float* __restrict__ y, float* __restrict__ e, int C) {
  int i = blockIdx.x * blockDim.x + threadIdx.x;
  if (i >= B_ * C) return;
  int b = i / C, c = i % C;
  const float* p = y + ((size_t)b * C + c) * N_;
  float m = p[0];
  for (int n = 1; n < N_; ++n) m = fmaxf(m, p[n]);
  e[i] = m;
}

// ---------------- MLP head --------------------------------------------------
__global__ void k_lin1(const float* __restrict__ e, const float* __restrict__ wl1,
                       float* __restrict__ h, int E, int J) {
  int i = blockIdx.x * blockDim.x + threadIdx.x;
  if (i >= B_ * J) return;
  int b = i / J, j = i % J;
  const float* ev = e + (size_t)b * E;
  const float* w = wl1 + (size_t)j * E;
  float s = 0.f;
  for (int k = 0; k < E; ++k) s += ev[k] * w[k];
  h[i] = s;
}

__global__ void k_bnb_gelu(float* __restrict__ h, const float* __restrict__ g,
                           const float* __restrict__ bb, int J) {
  int j = blockIdx.x * blockDim.x + threadIdx.x;
  if (j >= J) return;
  float a = h[j], c = h[J + j];                 // B_ == 2
  float m = 0.5f * (a + c);
  float var = 0.5f * ((a - m) * (a - m) + (c - m) * (c - m));
  float rs = rsqrtf(var + 1e-5f);
  h[j]     = gelu_f(g[j] * (a - m) * rs + bb[j]);
  h[J + j] = gelu_f(g[j] * (c - m) * rs + bb[j]);
}

__global__ void k_lin2(const float* __restrict__ h, const float* __restrict__ wl2,
                       const float* __restrict__ bl2, float* __restrict__ out,
                       int J, int O) {
  int i = blockIdx.x * blockDim.x + threadIdx.x;
  if (i >= B_ * O) return;
  int b = i / O, k = i % O;
  const float* hb = h + (size_t)b * J;
  const float* w = wl2 + (size_t)k * J;
  float s = bl2[k];
  for (int j = 0; j < J; ++j) s += hb[j] * w[j];
  out[i] = s;
}

// ===========================================================================
extern "C" void kernel_launch(void* const* d_in, const int* in_sizes, int n_in,
                              void* d_out, int out_size, void* d_ws, size_t ws_size,
                              hipStream_t stream) {
  (void)in_sizes; (void)n_in; (void)out_size; (void)ws_size;
  const float* x   = (const float*)d_in[0];
  const float* w1  = (const float*)d_in[1];
  const float* g1  = (const float*)d_in[2];
  const float* b1  = (const float*)d_in[3];
  const float* w2  = (const float*)d_in[4];
  const float* g2  = (const float*)d_in[5];
  const float* b2  = (const float*)d_in[6];
  const float* w3  = (const float*)d_in[7];
  const float* g3  = (const float*)d_in[8];
  const float* b3  = (const float*)d_in[9];
  const float* w4  = (const float*)d_in[10];
  const float* g4  = (const float*)d_in[11];
  const float* b4  = (const float*)d_in[12];
  const float* wl1 = (const float*)d_in[13];
  const float* g5  = (const float*)d_in[14];
  const float* b5  = (const float*)d_in[15];
  const float* wl2 = (const float*)d_in[16];
  const float* bl2 = (const float*)d_in[17];
  float* out = (float*)d_out;

  char* ws = (char*)d_ws;
  size_t off = 0;
  auto alloc = [&](size_t elems) -> float* {
    float* p = (float*)(ws + off);
    off += ((elems * sizeof(float) + 255) & ~(size_t)255);
    return p;
  };
  float* nd2   = alloc((size_t)B_ * N_ * N_);
  float* xx    = alloc((size_t)B_ * N_);
  int*   idx   = (int*)alloc((size_t)B_ * N_ * KNN_);
  float* nbuf  = alloc((size_t)B_ * 3 * N_);
  float* xpad  = alloc((size_t)B_ * 4 * N_);
  float* feat1 = alloc((size_t)B_ * 64 * N_);
  float* feat2 = alloc((size_t)B_ * 64 * N_);
  float* feat3 = alloc((size_t)B_ * 128 * N_);
  float* cat1  = alloc((size_t)B_ * 8 * N_);
  float* cat2  = alloc((size_t)B_ * 68 * N_);
  float* cat3  = alloc((size_t)B_ * 68 * N_);
  float* cat4  = alloc((size_t)B_ * 256 * N_);
  float* ybuf  = alloc((size_t)B_ * 1024 * N_);
  float* mu    = alloc(1024);
  float* rsig  = alloc(1024);
  float* ebuf  = alloc((size_t)B_ * 1024);
  float* hbuf  = alloc((size_t)B_ * 256);
  float* w1p   = alloc((size_t)64 * 8);
  float* w2p   = alloc((size_t)64 * 68);
  float* w3p   = alloc((size_t)128 * 68);

  auto zero = [&](float* p, int count) {
    k_zero<<<(count + 255) / 256, 256, 0, stream>>>(p, count);
  };
  auto pack = [&](float* dst, int dstC, int o, const float* src, int srcC, int nch) {
    int tot = B_ * nch * N_;
    k_pack<<<(tot + 255) / 256, 256, 0, stream>>>(dst, dstC, o, src, srcC, nch);
  };
  auto padw = [&](float* dst, const float* src, int O, int Cin, int CPad) {
    int tot = O * CPad;
    k_padw<<<(tot + 255) / 256, 256, 0, stream>>>(dst, src, O, Cin, CPad);
  };

  // ---- one-time-per-call padded operands ----
  zero(xpad, B_ * 4 * N_);
  pack(xpad, 4, 0, x, 3, 3);
  padw(w1p, w1, 64, 6, 8);
  padw(w2p, w2, 64, 67, 68);
  padw(w3p, w3, 128, 67, 68);

  const int GRAM_GRID = B_ * (N_ / 16) * (N_ / 64);

  // ---- stage 1 ----
  k_xx<<<(B_ * N_ + 255) / 256, 256, 0, stream>>>(xpad, xx, 4);
  k_gram_negd2<4><<<GRAM_GRID, 32, 0, stream>>>(xpad, xx, nd2);
  k_topk<<<(B_ * N_ + 255) / 256, 256, 0, stream>>>(nd2, idx);
  k_vecavg<<<B_ * N_, 256, 0, stream>>>(xpad, 4, idx, nbuf);
  zero(cat1, B_ * 8 * N_);
  pack(cat1, 8, 0, x, 3, 3);
  pack(cat1, 8, 3, nbuf, 3, 3);
  k_gemm_wmma<8><<<B_ * (64 / 16) * (N_ / 64), 32, 0, stream>>>(w1p, cat1, ybuf, 64);
  k_bnstats<<<64, 256, 0, stream>>>(ybuf, mu, rsig, 64);
  k_bngelu<<<(B_ * 64 * N_ + 255) / 256, 256, 0, stream>>>(ybuf, mu, rsig, g1, b1, feat1, 64);

  // ---- stage 2 ----
  k_xx<<<(B_ * N_ + 255) / 256, 256, 0, stream>>>(feat1, xx, 64);
  k_gram_negd2<64><<<GRAM_GRID, 32, 0, stream>>>(feat1, xx, nd2);
  k_topk<<<(B_ * N_ + 255) / 256, 256, 0, stream>>>(nd2, idx);
  k_vecavg<<<B_ * N_, 256, 0, stream>>>(feat1, 64, idx, nbuf);
  zero(cat2, B_ * 68 * N_);
  pack(cat2, 68, 0, feat1, 64, 64);
  pack(cat2, 68, 64, nbuf, 3, 3);
  k_gemm_wmma<68><<<B_ * (64 / 16) * (N_ / 64), 32, 0, stream>>>(w2p, cat2, ybuf, 64);
  k_bnstats<<<64, 256, 0, stream>>>(ybuf, mu, rsig, 64);
  k_bngelu<<<(B_ * 64 * N_ + 255) / 256, 256, 0, stream>>>(ybuf, mu, rsig, g2, b2, feat2, 64);

  // ---- stage 3 ----
  k_xx<<<(B_ * N_ + 255) / 256, 256, 0, stream>>>(feat2, xx, 64);
  k_gram_negd2<64><<<GRAM_GRID, 32, 0, stream>>>(feat2, xx, nd2);
  k_topk<<<(B_ * N_ + 255) / 256, 256, 0, stream>>>(nd2, idx);
  k_vecavg<<<B_ * N_, 256, 0, stream>>>(feat2, 64, idx, nbuf);
  zero(cat3, B_ * 68 * N_);
  pack(cat3, 68, 0, feat2, 64, 64);
  pack(cat3, 68, 64, nbuf, 3, 3);
  k_gemm_wmma<68><<<B_ * (128 / 16) * (N_ / 64), 32, 0, stream>>>(w3p, cat3, ybuf, 128);
  k_bnstats<<<128, 256, 0, stream>>>(ybuf, mu, rsig, 128);
  k_bngelu<<<(B_ * 128 * N_ + 255) / 256, 256, 0, stream>>>(ybuf, mu, rsig, g3, b3, feat3, 128);

  // ---- stage 4: conv over cat(x1,x2,x3) ----
  pack(cat4, 256, 0, feat1, 64, 64);
  pack(cat4, 256, 64, feat2, 64, 64);
  pack(cat4, 256, 128, feat3, 128, 128);
  k_gemm_wmma<256><<<B_ * (1024 / 16) * (N_ / 64), 32, 0, stream>>>(w4, cat4, ybuf, 1024);
  k_bnstats<<<1024, 256, 0, stream>>>(ybuf, mu, rsig, 1024);
  k_bngelu<<<(B_ * 1024 * N_ + 255) / 256, 256, 0, stream>>>(ybuf, mu, rsig, g4, b4, ybuf, 1024);

  // ---- head ----
  k_maxpool<<<(B_ * 1024 + 255) / 256, 256, 0, stream>>>(ybuf, ebuf, 1024);
  k_lin1<<<(B_ * 256 + 255) / 256, 256, 0, stream>>>(ebuf, wl1, hbuf, 1024, 256);
  k_bnb_gelu<<<1, 256, 0, stream>>>(hbuf, g5, b5, 256);
  k_lin2<<<1, 128, 0, stream>>>(hbuf, wl2, bl2, out, 256, 40);
}